// GATClassifier_17154099380448
// MI455X (gfx1250) — compile-verified
//
#include <hip/hip_runtime.h>

// ---------------- problem constants (match reference) ----------------
#define N_NODES 50000
#define NPAD    50048   // node count padded to multiple of 64 (workspace only)
#define N_EDGES 800000
#define HID     64
#define HEADS   4
#define NBATCH  256
#define TEXT    768
#define VIS     512
#define AUD     128

typedef float v2f __attribute__((ext_vector_type(2)));
typedef float v8f __attribute__((ext_vector_type(8)));

// f32 WMMA: D(16x16) = A(16x4) * B(4x16) + C   (CDNA5 V_WMMA_F32_16X16X4_F32)
__device__ __forceinline__ v8f wmma_f32(v2f a, v2f b, v8f c) {
  return __builtin_amdgcn_wmma_f32_16x16x4_f32(
      /*neg_a=*/false, a, /*neg_b=*/false, b,
      /*c_mod=*/(short)0, c, /*reuse_a=*/false, /*reuse_b=*/false);
}

// monotonic float->uint key for atomicMax-based segment max (handles negatives)
__device__ __forceinline__ unsigned fkey(float f) {
  unsigned u = __float_as_uint(f);
  return (u & 0x80000000u) ? ~u : (u | 0x80000000u);
}
__device__ __forceinline__ float funkey(unsigned k) {
  unsigned u = (k & 0x80000000u) ? (k & 0x7fffffffu) : ~k;
  return __uint_as_float(u);
}

// ---------------- zero fill ----------------
__global__ void zero_kernel(float* __restrict__ p, long long n) {
  long long t = (long long)blockIdx.x * blockDim.x + threadIdx.x;
  if (t < n) p[t] = 0.0f;
}

// ---------------- fused typed input projection: h0 = select(x@Wt, x@Wv, x@Wa) ----------------
// one wave per 16-row tile; three branch-free K phases so accumulators stay pinned
__global__ __launch_bounds__(32) void proj_kernel(
    const float* __restrict__ x, const int* __restrict__ ntype,
    const float* __restrict__ Wt, const float* __restrict__ bt,
    const float* __restrict__ Wv, const float* __restrict__ bv,
    const float* __restrict__ Wa, const float* __restrict__ ba,
    float* __restrict__ h0)
{
  const int m0   = blockIdx.x * 16;
  const int lane = threadIdx.x;
  const int half = lane >> 4;     // 0: K lo pair / M 0..7 ; 1: K hi pair / M 8..15
  const int r    = lane & 15;

  v8f acct[4], accv[4], acca[4];
#pragma unroll
  for (int i = 0; i < 4; i++) { acct[i] = v8f{}; accv[i] = v8f{}; acca[i] = v8f{}; }

  const float* xrow = x + (size_t)(m0 + r) * TEXT;

  // phase A: k in [0,128): text + vision + audio
  for (int k = 0; k < AUD; k += 4) {
    const int kb = k + 2 * half;
    v2f a = *(const v2f*)(xrow + kb);
#pragma unroll
    for (int nt = 0; nt < 4; nt++) {
      const int col = nt * 16 + r;
      v2f bw;  bw.x  = Wt[(size_t)(kb + 0) * HID + col]; bw.y  = Wt[(size_t)(kb + 1) * HID + col];
      v2f bw2; bw2.x = Wv[(size_t)(kb + 0) * HID + col]; bw2.y = Wv[(size_t)(kb + 1) * HID + col];
      v2f bw3; bw3.x = Wa[(size_t)(kb + 0) * HID + col]; bw3.y = Wa[(size_t)(kb + 1) * HID + col];
      acct[nt] = wmma_f32(a, bw,  acct[nt]);
      accv[nt] = wmma_f32(a, bw2, accv[nt]);
      acca[nt] = wmma_f32(a, bw3, acca[nt]);
    }
  }
  // phase B: k in [128,512): text + vision
  for (int k = AUD; k < VIS; k += 4) {
    const int kb = k + 2 * half;
    v2f a = *(const v2f*)(xrow + kb);
#pragma unroll
    for (int nt = 0; nt < 4; nt++) {
      const int col = nt * 16 + r;
      v2f bw;  bw.x  = Wt[(size_t)(kb + 0) * HID + col]; bw.y  = Wt[(size_t)(kb + 1) * HID + col];
      v2f bw2; bw2.x = Wv[(size_t)(kb + 0) * HID + col]; bw2.y = Wv[(size_t)(kb + 1) * HID + col];
      acct[nt] = wmma_f32(a, bw,  acct[nt]);
      accv[nt] = wmma_f32(a, bw2, accv[nt]);
    }
  }
  // phase C: k in [512,768): text only
  for (int k = VIS; k < TEXT; k += 4) {
    const int kb = k + 2 * half;
    v2f a = *(const v2f*)(xrow + kb);
#pragma unroll
    for (int nt = 0; nt < 4; nt++) {
      const int col = nt * 16 + r;
      v2f bw; bw.x = Wt[(size_t)(kb + 0) * HID + col]; bw.y = Wt[(size_t)(kb + 1) * HID + col];
      acct[nt] = wmma_f32(a, bw, acct[nt]);
    }
  }

  // C layout: element i of v8f -> row m0 + i + 8*half, col = ntile*16 + r
#pragma unroll
  for (int nt = 0; nt < 4; nt++) {
    const int col = nt * 16 + r;
#pragma unroll
    for (int i = 0; i < 8; i++) {
      const int m = m0 + i + 8 * half;
      const int ty = ntype[m];
      float val = (ty == 0) ? (acct[nt][i] + bt[col])
                : (ty == 1) ? (accv[nt][i] + bv[col])
                            : (acca[nt][i] + ba[col]);
      h0[(size_t)m * HID + col] = val;
    }
  }
}

// ---------------- GEMM: 64x64 tile per wave (4 M-tiles x 4 N-tiles, 16 WMMA/K-step) ----------------
// A rows must be readable up to NPAD (workspace-resident); pad rows produce garbage that
// lands in pad rows of C and is never consumed.
__global__ __launch_bounds__(32) void gemm_kernel(
    const float* __restrict__ A, const float* __restrict__ Bm,
    float* __restrict__ C, int K, int Nout)
{
  const int m0   = blockIdx.x * 64;
  const int n0   = blockIdx.y * 64;
  const int lane = threadIdx.x;
  const int half = lane >> 4;
  const int r    = lane & 15;

  v8f acc[4][4];
#pragma unroll
  for (int mt = 0; mt < 4; mt++)
#pragma unroll
    for (int nt = 0; nt < 4; nt++) acc[mt][nt] = v8f{};

  const float* arow[4];
#pragma unroll
  for (int mt = 0; mt < 4; mt++)
    arow[mt] = A + (size_t)(m0 + mt * 16 + r) * K;

  for (int k = 0; k < K; k += 4) {
    const int kb = k + 2 * half;
    v2f a[4];
#pragma unroll
    for (int mt = 0; mt < 4; mt++) a[mt] = *(const v2f*)(arow[mt] + kb);
    v2f b[4];
#pragma unroll
    for (int nt = 0; nt < 4; nt++) {
      const int col = n0 + nt * 16 + r;
      b[nt].x = Bm[(size_t)(kb + 0) * Nout + col];
      b[nt].y = Bm[(size_t)(kb + 1) * Nout + col];
    }
#pragma unroll
    for (int mt = 0; mt < 4; mt++)
#pragma unroll
      for (int nt = 0; nt < 4; nt++)
        acc[mt][nt] = wmma_f32(a[mt], b[nt], acc[mt][nt]);
  }

#pragma unroll
  for (int mt = 0; mt < 4; mt++)
#pragma unroll
    for (int nt = 0; nt < 4; nt++) {
      const int col = n0 + nt * 16 + r;
#pragma unroll
      for (int i = 0; i < 8; i++)
        C[(size_t)(m0 + mt * 16 + i + 8 * half) * Nout + col] = acc[mt][nt][i];
    }
}

// ---------------- attention logits: as[n,h] = <h[n,h,:], a_src[h,:]>, ad likewise ----------------
__global__ void alpha_kernel(const float* __restrict__ h,
                             const float* __restrict__ asrc,
                             const float* __restrict__ adst,
                             float* __restrict__ as_, float* __restrict__ ad_,
                             int heads)
{
  int t = blockIdx.x * blockDim.x + threadIdx.x;
  if (t >= N_NODES * heads) return;
  const int n = t / heads, hh = t % heads;
  const float* hv = h + (size_t)n * heads * HID + hh * HID;
  const float* s = asrc + hh * HID;
  const float* d = adst + hh * HID;
  float ss = 0.f, dd = 0.f;
  for (int c = 0; c < HID; c++) { float v = hv[c]; ss += v * s[c]; dd += v * d[c]; }
  as_[t] = ss; ad_[t] = dd;
}

// ---------------- pass A: segment max over dst (atomicMax with monotonic key) ----------------
__global__ void edge_max_kernel(const int* __restrict__ ei,
                                const float* __restrict__ as_,
                                const float* __restrict__ ad_,
                                unsigned* __restrict__ mkey, int heads)
{
  int t = blockIdx.x * blockDim.x + threadIdx.x;
  const int ne = N_EDGES + N_NODES;
  if (t >= ne * heads) return;
  const int e = t / heads, hh = t % heads;
  int s, d;
  if (e < N_EDGES) { s = ei[e]; d = ei[N_EDGES + e]; } else { s = d = e - N_EDGES; }
  float v = as_[s * heads + hh] + ad_[d * heads + hh];
  v = (v > 0.f) ? v : 0.2f * v;            // leaky_relu(0.2)
  atomicMax(&mkey[d * heads + hh], fkey(v));
}

// ---------------- pass B: denom[d,h] += exp(lrelu - max) ----------------
__global__ void edge_den_kernel(const int* __restrict__ ei,
                                const float* __restrict__ as_,
                                const float* __restrict__ ad_,
                                const unsigned* __restrict__ mkey,
                                float* __restrict__ den, int heads)
{
  int t = blockIdx.x * blockDim.x + threadIdx.x;
  const int ne = N_EDGES + N_NODES;
  if (t >= ne * heads) return;
  const int e = t / heads, hh = t % heads;
  int s, d;
  if (e < N_EDGES) { s = ei[e]; d = ei[N_EDGES + e]; } else { s = d = e - N_EDGES; }
  float v = as_[s * heads + hh] + ad_[d * heads + hh];
  v = (v > 0.f) ? v : 0.2f * v;
  atomicAdd(&den[d * heads + hh], __expf(v - funkey(mkey[d * heads + hh])));
}

// ---------------- pass C: agg[d,:] += exp(...) * h[s,:]  (32 lanes per edge) ----------------
__global__ void edge_agg_kernel(const int* __restrict__ ei,
                                const float* __restrict__ as_,
                                const float* __restrict__ ad_,
                                const unsigned* __restrict__ mkey,
                                const float* __restrict__ h,
                                float* __restrict__ agg, int heads)
{
  long long t = (long long)blockIdx.x * blockDim.x + threadIdx.x;
  const long long ne = N_EDGES + N_NODES;
  long long e = t >> 5;
  if (e >= ne) return;
  const int lane = (int)(t & 31);
  int s, d;
  if (e < N_EDGES) { s = ei[e]; d = ei[N_EDGES + e]; } else { s = d = (int)(e - N_EDGES); }
  const int vals  = heads * 2;             // (heads*64)/32 values per lane
  const int jbase = lane * vals;
  for (int j = 0; j < vals; j++) {
    const int jj = jbase + j;
    const int hh = jj >> 6;
    float v = as_[s * heads + hh] + ad_[d * heads + hh];
    v = (v > 0.f) ? v : 0.2f * v;
    const float ex = __expf(v - funkey(mkey[d * heads + hh]));
    atomicAdd(&agg[(size_t)d * heads * HID + jj],
              ex * h[(size_t)s * heads * HID + jj]);
  }
}

// ---------------- layer-1 epilogue: h1e = elu(agg/denom + b1) (in place) ----------------
__global__ void epi1_kernel(float* __restrict__ agg, const float* __restrict__ den,
                            const float* __restrict__ b1)
{
  int t = blockIdx.x * blockDim.x + threadIdx.x;
  if (t >= N_NODES * HEADS * HID) return;
  const int n = t >> 8, j = t & 255;
  float v = agg[t] / (den[n * HEADS + (j >> 6)] + 1e-16f) + b1[j];
  agg[t] = (v > 0.f) ? v : (__expf(v) - 1.f);
}

// ---------------- layer-2 epilogue + batch pooling ----------------
__global__ void epi2_pool_kernel(const float* __restrict__ agg2,
                                 const float* __restrict__ den2,
                                 const float* __restrict__ b2,
                                 const int* __restrict__ batch,
                                 float* __restrict__ pool, float* __restrict__ cnt)
{
  int t = blockIdx.x * blockDim.x + threadIdx.x;
  if (t >= N_NODES * HID) return;
  const int n = t >> 6, c = t & 63;
  const float v = agg2[t] / (den2[n] + 1e-16f) + b2[c];
  const int b = batch[n];
  atomicAdd(&pool[(size_t)b * HID + c], v);
  if (c == 0) atomicAdd(&cnt[b], 1.0f);
}

// ---------------- readout: out[b] = (pool[b]/max(cnt,1)) @ Wl + bl ----------------
__global__ void final_kernel(const float* __restrict__ pool, const float* __restrict__ cnt,
                             const float* __restrict__ Wl, const float* __restrict__ bl,
                             float* __restrict__ out)
{
  int b = blockIdx.x * blockDim.x + threadIdx.x;
  if (b >= NBATCH) return;
  float c = cnt[b]; c = (c > 1.f) ? c : 1.f;
  float acc = 0.f;
  for (int j = 0; j < HID; j++) acc += (pool[(size_t)b * HID + j] / c) * Wl[j];
  out[b] = acc + bl[0];
}

static inline int cdiv(long long a, int b) { return (int)((a + b - 1) / b); }

extern "C" void kernel_launch(void* const* d_in, const int* in_sizes, int n_in,
                              void* d_out, int out_size, void* d_ws, size_t ws_size,
                              hipStream_t stream) {
  (void)in_sizes; (void)n_in; (void)out_size; (void)ws_size;
  const float* x     = (const float*)d_in[0];
  const int*   ntype = (const int*)  d_in[1];
  const int*   ei    = (const int*)  d_in[2];
  const int*   batch = (const int*)  d_in[3];
  const float* Wt = (const float*)d_in[4];  const float* bt = (const float*)d_in[5];
  const float* Wv = (const float*)d_in[6];  const float* bv = (const float*)d_in[7];
  const float* Wa = (const float*)d_in[8];  const float* ba = (const float*)d_in[9];
  const float* W1 = (const float*)d_in[10];
  const float* a1s = (const float*)d_in[11]; const float* a1d = (const float*)d_in[12];
  const float* b1  = (const float*)d_in[13];
  const float* W2 = (const float*)d_in[14];
  const float* a2s = (const float*)d_in[15]; const float* a2d = (const float*)d_in[16];
  const float* b2  = (const float*)d_in[17];
  const float* Wl = (const float*)d_in[18]; const float* bl = (const float*)d_in[19];
  float* out = (float*)d_out;

  // workspace layout (floats); matrix buffers padded to NPAD rows
  float* ws = (float*)d_ws;
  const size_t NN = N_NODES;
  float*    h0   = ws;                               // NPAD*64  (also h2lin in layer 2)
  float*    h1   = h0   + (size_t)NPAD * 64;         // NPAD*256 (first N*64 reused as agg2)
  float*    agg1 = h1   + (size_t)NPAD * 256;        // NPAD*256 (becomes h1e after epi1)
  float*    as1  = agg1 + (size_t)NPAD * 256;        // N*4
  float*    ad1  = as1  + NN * 4;                    // N*4
  unsigned* mk1  = (unsigned*)(ad1 + NN * 4);        // N*4
  float*    den1 = (float*)(mk1 + NN * 4);           // N*4
  float*    as2  = den1 + NN * 4;                    // N
  float*    ad2  = as2  + NN;                        // N
  unsigned* mk2  = (unsigned*)(ad2 + NN);            // N
  float*    den2 = (float*)(mk2 + NN);               // N
  float*    pool = den2 + NN;                        // B*64
  float*    cnt  = pool + (size_t)NBATCH * 64;       // B
  float*    agg2 = h1;                               // reuse (after layer-1 agg done)

  const int NE = N_EDGES + N_NODES;

  // ---- zero the accumulators used this call ----
  zero_kernel<<<cdiv(NN * 4, 256), 256, 0, stream>>>((float*)mk1, NN * 4);
  zero_kernel<<<cdiv(NN * 4, 256), 256, 0, stream>>>(den1, NN * 4);
  zero_kernel<<<cdiv(NN * 256, 256), 256, 0, stream>>>(agg1, NN * 256);
  zero_kernel<<<cdiv(NN, 256), 256, 0, stream>>>((float*)mk2, NN);
  zero_kernel<<<cdiv(NN, 256), 256, 0, stream>>>(den2, NN);
  zero_kernel<<<cdiv((size_t)NBATCH * 65, 256), 256, 0, stream>>>(pool, (size_t)NBATCH * 65);

  // ---- typed projection (WMMA, exact 3125 tiles, no OOB on x) ----
  proj_kernel<<<N_NODES / 16, 32, 0, stream>>>(x, ntype, Wt, bt, Wv, bv, Wa, ba, h0);

  // ---- GAT layer 1 ----
  gemm_kernel<<<dim3(NPAD / 64, 4), 32, 0, stream>>>(h0, W1, h1, 64, 256);
  alpha_kernel<<<cdiv((long long)N_NODES * HEADS, 256), 256, 0, stream>>>(h1, a1s, a1d, as1, ad1, HEADS);
  edge_max_kernel<<<cdiv((long long)NE * HEADS, 256), 256, 0, stream>>>(ei, as1, ad1, mk1, HEADS);
  edge_den_kernel<<<cdiv((long long)NE * HEADS, 256), 256, 0, stream>>>(ei, as1, ad1, mk1, den1, HEADS);
  edge_agg_kernel<<<cdiv((long long)NE * 32, 256), 256, 0, stream>>>(ei, as1, ad1, mk1, h1, agg1, HEADS);
  epi1_kernel<<<cdiv((long long)N_NODES * 256, 256), 256, 0, stream>>>(agg1, den1, b1);

  // ---- GAT layer 2 ----
  gemm_kernel<<<dim3(NPAD / 64, 1), 32, 0, stream>>>(agg1, W2, h0, 256, 64);
  zero_kernel<<<cdiv(NN * 64, 256), 256, 0, stream>>>(agg2, NN * 64);
  alpha_kernel<<<cdiv((long long)N_NODES, 256), 256, 0, stream>>>(h0, a2s, a2d, as2, ad2, 1);
  edge_max_kernel<<<cdiv((long long)NE, 256), 256, 0, stream>>>(ei, as2, ad2, mk2, 1);
  edge_den_kernel<<<cdiv((long long)NE, 256), 256, 0, stream>>>(ei, as2, ad2, mk2, den2, 1);
  edge_agg_kernel<<<cdiv((long long)NE * 32, 256), 256, 0, stream>>>(ei, as2, ad2, mk2, h0, agg2, 1);
  epi2_pool_kernel<<<cdiv((long long)N_NODES * 64, 256), 256, 0, stream>>>(agg2, den2, b2, batch, pool, cnt);

  // ---- readout ----
  final_kernel<<<1, 256, 0, stream>>>(pool, cnt, Wl, bl, out);
}